// GraphNCA_67061619360164
// MI455X (gfx1250) — compile-verified
//
#include <hip/hip_runtime.h>
#include <hip/hip_bf16.h>

typedef float v2f __attribute__((ext_vector_type(2)));
typedef float v8f __attribute__((ext_vector_type(8)));

// D(16x16,f32) = A(16x4,f32) * B(4x16,f32) + C
// 8-arg form: (neg_a, A, neg_b, B, c_mod, C, reuse_a, reuse_b)
#define WMMA_F32X4(a, b, c) \
  __builtin_amdgcn_wmma_f32_16x16x4_f32(false, (a), false, (b), (short)0, (c), false, false)

// Fire-and-forget hardware f32 atomic add (global_atomic_add_f32, no return,
// STOREcnt-tracked) — avoids any CAS-loop lowering of plain atomicAdd(float*).
__device__ __forceinline__ void atomic_add_f32(float* p, float v) {
  __hip_atomic_fetch_add(p, v, __ATOMIC_RELAXED, __HIP_MEMORY_SCOPE_AGENT);
}

// ---------------------------------------------------------------- degree prep
__global__ void __launch_bounds__(256) k_init_deg(float* deg, int n) {
  int i = blockIdx.x * 256 + threadIdx.x;
  if (i < n) deg[i] = 1.0f;  // self-loop contributes 1
}

__global__ void __launch_bounds__(256) k_deg_accum(const int* __restrict__ col,
                                                   float* deg, int e) {
  int i = blockIdx.x * 256 + threadIdx.x;
  if (i < e) atomic_add_f32(&deg[col[i]], 1.0f);
}

__global__ void __launch_bounds__(256) k_deg_to_dis(float* deg, int n) {
  int i = blockIdx.x * 256 + threadIdx.x;
  if (i < n) deg[i] = rsqrtf(deg[i]);  // deg >= 1 always (self-loops)
}

// ---------------------------------------------------------------- h = xx @ W_perc   [N,16]x[16,48]
__global__ void __launch_bounds__(256) k_xw_wmma(const float* __restrict__ xx,
                                                 const float* __restrict__ Wp,
                                                 float* __restrict__ h, int n) {
  int wave    = (blockIdx.x * 256 + threadIdx.x) >> 5;
  int lane    = threadIdx.x & 31;
  int rowBase = wave * 16;
  if (rowBase >= n) return;             // wave-uniform: EXEC stays all-ones for WMMA
  int half = lane >> 4;
  int r    = lane & 15;

  v8f c0 = {}, c1 = {}, c2 = {};
  const float* xrow = xx + (size_t)(rowBase + r) * 16;
#pragma unroll
  for (int kb = 0; kb < 16; kb += 4) {
    int k0 = kb + 2 * half;
    v2f a;  a.x = xrow[k0];              a.y = xrow[k0 + 1];
    v2f b0; b0.x = Wp[k0 * 48 +      r]; b0.y = Wp[(k0 + 1) * 48 +      r];
    v2f b1; b1.x = Wp[k0 * 48 + 16 + r]; b1.y = Wp[(k0 + 1) * 48 + 16 + r];
    v2f b2; b2.x = Wp[k0 * 48 + 32 + r]; b2.y = Wp[(k0 + 1) * 48 + 32 + r];
    c0 = WMMA_F32X4(a, b0, c0);
    c1 = WMMA_F32X4(a, b1, c1);
    c2 = WMMA_F32X4(a, b2, c2);
  }
#pragma unroll
  for (int v = 0; v < 8; v++) {
    int m = v + 8 * half;                // C/D layout: vgpr v, lane-half -> row
    float* o = h + (size_t)(rowBase + m) * 48;
    o[r] = c0[v]; o[16 + r] = c1[v]; o[32 + r] = c2[v];
  }
}

// ---------------------------------------------------------------- feat = h * dis^2 (self loop)
__global__ void __launch_bounds__(256) k_feat_init(const float* __restrict__ h,
                                                   const float* __restrict__ dis,
                                                   float* __restrict__ feat, long total) {
  long t = (long)blockIdx.x * 256 + threadIdx.x;
  if (t >= total) return;
  int node = (int)(t / 48);
  float d = dis[node];
  feat[t] = h[t] * d * d;
}

// ---------------------------------------------------------------- edge scatter-add
__global__ void __launch_bounds__(256) k_edge_scatter(const int* __restrict__ src,
                                                      const int* __restrict__ dst,
                                                      const float* __restrict__ h,
                                                      const float* __restrict__ dis,
                                                      float* __restrict__ feat, long total) {
  long t = (long)blockIdx.x * 256 + threadIdx.x;
  if (t >= total) return;
  int e = (int)(t / 48);
  int k = (int)(t - (long)e * 48);
  int rs = src[e], cs = dst[e];
  float nrm = dis[rs] * dis[cs];
  atomic_add_f32(&feat[(size_t)cs * 48 + k], h[(size_t)rs * 48 + k] * nrm);
}

// ---------------------------------------------------------------- MLP: xout = xx + relu(feat@W1+b1)@W2+b2
__global__ void __launch_bounds__(256) k_mlp_wmma(const float* __restrict__ feat,
                                                  const float* __restrict__ xx,
                                                  const float* __restrict__ W1,
                                                  const float* __restrict__ b1,
                                                  const float* __restrict__ W2,
                                                  const float* __restrict__ b2,
                                                  float* __restrict__ xout, int n) {
  __shared__ float lds[8][16 * 32];      // per-wave staging of hidden tile
  int waveInBlk = threadIdx.x >> 5;
  int wave      = blockIdx.x * 8 + waveInBlk;
  int lane      = threadIdx.x & 31;
  int rowBase   = wave * 16;
  if (rowBase >= n) return;
  int half = lane >> 4;
  int r    = lane & 15;

  // GEMM1: [16,48] x [48,32] -> hidden [16,32]
  v8f h0 = {}, h1 = {};
#pragma unroll
  for (int kb = 0; kb < 48; kb += 4) {
    int k0 = kb + 2 * half;
    const float* frow = feat + (size_t)(rowBase + r) * 48;
    v2f a;  a.x = frow[k0];               a.y = frow[k0 + 1];
    v2f b0; b0.x = W1[k0 * 32 +      r];  b0.y = W1[(k0 + 1) * 32 +      r];
    v2f b1v; b1v.x = W1[k0 * 32 + 16 + r]; b1v.y = W1[(k0 + 1) * 32 + 16 + r];
    h0 = WMMA_F32X4(a, b0, h0);
    h1 = WMMA_F32X4(a, b1v, h1);
  }
  float bia0 = b1[r], bia1 = b1[16 + r];
  float* my = lds[waveInBlk];
#pragma unroll
  for (int v = 0; v < 8; v++) {
    int m = v + 8 * half;
    float t0 = h0[v] + bia0; t0 = t0 > 0.0f ? t0 : 0.0f;
    float t1 = h1[v] + bia1; t1 = t1 > 0.0f ? t1 : 0.0f;
    my[m * 32 + r]      = t0;
    my[m * 32 + 16 + r] = t1;
  }
  asm volatile("s_wait_dscnt 0" ::: "memory");  // LDS in-order per wave; wait before re-read

  // GEMM2: hidden [16,32] x [32,16] -> update [16,16]
  v8f u = {};
#pragma unroll
  for (int kb = 0; kb < 32; kb += 4) {
    int k0 = kb + 2 * half;
    v2f a;  a.x = my[r * 32 + k0];       a.y = my[r * 32 + k0 + 1];
    v2f bb; bb.x = W2[k0 * 16 + r];      bb.y = W2[(k0 + 1) * 16 + r];
    u = WMMA_F32X4(a, bb, u);
  }
  float bia2 = b2[r];
#pragma unroll
  for (int v = 0; v < 8; v++) {
    int m = v + 8 * half;
    size_t idx = (size_t)(rowBase + m) * 16 + r;
    xout[idx] = xx[idx] + u[v] + bia2;
  }
}

// ---------------------------------------------------------------- parent split (tiny)
__global__ void k_parent_split(const float* __restrict__ xout, const int* __restrict__ pidx,
                               const float* __restrict__ S1, const float* __restrict__ sb1,
                               const float* __restrict__ S2, const float* __restrict__ sb2,
                               float* __restrict__ out_d) {
  __shared__ float p[16];
  __shared__ float hid[32];
  int lane = threadIdx.x;                // 32 threads
  int parent = pidx[0];
  if (lane < 16) p[lane] = xout[(size_t)parent * 16 + lane];
  __syncthreads();
  float acc = sb1[lane];
#pragma unroll
  for (int k = 0; k < 16; k++) acc += p[k] * S1[k * 32 + lane];
  hid[lane] = acc > 0.0f ? acc : 0.0f;
  __syncthreads();
  float o = sb2[lane];
#pragma unroll
  for (int j = 0; j < 32; j++) o += hid[j] * S2[j * 32 + lane];
  out_d[lane] = o;                       // daughters = split[0:32] flat
}

// ---------------------------------------------------------------- launcher
extern "C" void kernel_launch(void* const* d_in, const int* in_sizes, int n_in,
                              void* d_out, int out_size, void* d_ws, size_t ws_size,
                              hipStream_t stream) {
  const float* xx   = (const float*)d_in[0];
  const int*   edge = (const int*)d_in[1];
  const int*   pidx = (const int*)d_in[2];
  const float* Wp   = (const float*)d_in[3];
  const float* W1   = (const float*)d_in[4];
  const float* b1   = (const float*)d_in[5];
  const float* W2   = (const float*)d_in[6];
  const float* b2   = (const float*)d_in[7];
  const float* S1   = (const float*)d_in[8];
  const float* sb1  = (const float*)d_in[9];
  const float* S2   = (const float*)d_in[10];
  const float* sb2  = (const float*)d_in[11];

  const int N = in_sizes[0] / 16;        // 100000
  const int E = in_sizes[1] / 2;         // 3200000
  const int* src = edge;                 // edge_index[0]
  const int* dst = edge + E;             // edge_index[1]

  float* xout = (float*)d_out;           // [N,16]
  float* daughters = xout + (size_t)N * 16;

  float* dis  = (float*)d_ws;            // N floats (deg -> dis in place)
  float* h    = dis + N;                 // N*48
  float* feat = h + (size_t)N * 48;      // N*48

  const long totNC = (long)N * 48;
  const long totEC = (long)E * 48;
  const int  tiles = (N + 15) / 16;
  const int  gemmBlocks = (tiles + 7) / 8;   // 8 waves (16-row tiles) per 256-thread block

  k_init_deg   <<<(N + 255) / 256, 256, 0, stream>>>(dis, N);
  k_deg_accum  <<<(E + 255) / 256, 256, 0, stream>>>(dst, dis, E);
  k_deg_to_dis <<<(N + 255) / 256, 256, 0, stream>>>(dis, N);
  k_xw_wmma    <<<gemmBlocks, 256, 0, stream>>>(xx, Wp, h, N);
  k_feat_init  <<<(int)((totNC + 255) / 256), 256, 0, stream>>>(h, dis, feat, totNC);
  k_edge_scatter<<<(int)((totEC + 255) / 256), 256, 0, stream>>>(src, dst, h, dis, feat, totEC);
  k_mlp_wmma   <<<gemmBlocks, 256, 0, stream>>>(feat, xx, W1, b1, W2, b2, xout, N);
  k_parent_split<<<1, 32, 0, stream>>>(xout, pidx, S1, sb1, S2, sb2, daughters);
}